// SSMCore_50818053046523
// MI455X (gfx1250) — compile-verified
//
#include <hip/hip_runtime.h>

// SSM core for MI455X (gfx1250, wave32).
// ws layout (floats): xc[T*D] | proj[T*32] | dt[T*D]   (~33 MB total)

typedef __attribute__((ext_vector_type(2))) float v2f;
typedef __attribute__((ext_vector_type(8))) float v8f;
typedef __attribute__((ext_vector_type(4))) int v4i;

#define AS1 __attribute__((address_space(1)))
#define AS3 __attribute__((address_space(3)))

constexpr int T = 2048;
constexpr int D = 2048;
constexpr int S = 16;
constexpr int CONVW = 4;
constexpr float TINYV = 1e-38f;

__device__ __forceinline__ float softplusf(float v) {
  return (v > 20.0f) ? v : log1pf(__expf(v));
}

// ---- gfx1250 async global->LDS copy (ASYNCcnt-tracked), with safe fallback --
#if __has_builtin(__builtin_amdgcn_global_load_async_to_lds_b128)
#define ASYNC_LDS 1
#else
#define ASYNC_LDS 0
#endif

__device__ __forceinline__ void cp16(const float* g, float* l) {
#if ASYNC_LDS
  // params are (v4i AS1*, v4i AS3*, imm offset, imm cpol) per clang diagnostic
  __builtin_amdgcn_global_load_async_to_lds_b128(
      (AS1 v4i*)(g), (AS3 v4i*)(l), 0, 0);
#else
  *(float4*)l = *(const float4*)g;
#endif
}

__device__ __forceinline__ void cp_wait_all() {
#if ASYNC_LDS
#if __has_builtin(__builtin_amdgcn_s_wait_asynccnt)
  __builtin_amdgcn_s_wait_asynccnt(0);
#else
  asm volatile("s_wait_asynccnt 0x0" ::: "memory");
#endif
#endif
}

// ---------------------------------------------------------------------------
// Kernel 1: depthwise causal conv1d (width 4) + bias + SiLU  -> xc (T, D)
// ---------------------------------------------------------------------------
__global__ void k_conv_silu(const float* __restrict__ x,
                            const float* __restrict__ cw,   // (D, 1, 4)
                            const float* __restrict__ cb,   // (D)
                            float* __restrict__ xc) {
  const int idx = blockIdx.x * blockDim.x + threadIdx.x;    // t*D + d
  const int t = idx >> 11;
  const int d = idx & (D - 1);
  float acc = cb[d];
#pragma unroll
  for (int k = 0; k < CONVW; ++k) {
    const int tt = t - (CONVW - 1) + k;
    if (tt >= 0) acc = fmaf(x[tt * D + d], cw[d * CONVW + k], acc);
  }
  const float sig = 1.0f / (1.0f + __expf(-acc));
  xc[idx] = acc * sig;
}

// ---------------------------------------------------------------------------
// Kernel 2: proj = xc @ xproj_w^T   (M=T, N=32, K=D) via V_WMMA_F32_16X16X4_F32
// One wave per 16x16 tile; K=2048 over two accumulators for ILP.
// ---------------------------------------------------------------------------
__global__ void k_proj_wmma(const float* __restrict__ xc,   // (T, D)
                            const float* __restrict__ w,    // (32, D)
                            float* __restrict__ proj) {     // (T, 32)
  const int lane = threadIdx.x & 31;
  const int waveId = blockIdx.x * (blockDim.x >> 5) + (threadIdx.x >> 5); // 0..255
  const int m0 = (waveId >> 1) * 16;
  const int n0 = (waveId & 1) * 16;
  const int l16 = lane & 15;
  const int koff = (lane >> 4) << 1;      // 0 or 2

  const float* arow = xc + (m0 + l16) * D;
  const float* brow = w + (n0 + l16) * D;

  v8f c0 = {};
  v8f c1 = {};
#pragma unroll 4
  for (int k0 = 0; k0 < D; k0 += 8) {
    v2f a0 = *(const v2f*)(arow + k0 + koff);
    v2f b0 = *(const v2f*)(brow + k0 + koff);
    c0 = __builtin_amdgcn_wmma_f32_16x16x4_f32(false, a0, false, b0,
                                               (short)0, c0, false, false);
    v2f a1 = *(const v2f*)(arow + k0 + 4 + koff);
    v2f b1 = *(const v2f*)(brow + k0 + 4 + koff);
    c1 = __builtin_amdgcn_wmma_f32_16x16x4_f32(false, a1, false, b1,
                                               (short)0, c1, false, false);
  }

  const int col = n0 + l16;
  const int rbase = m0 + (lane >> 4) * 8;
#pragma unroll
  for (int r = 0; r < 8; ++r) {
    proj[(rbase + r) * (2 * S) + col] = c0[r] + c1[r];
  }
}

// ---------------------------------------------------------------------------
// Kernel 3: dt = softplus(softplus(B_ssm @ dt_w^T + dt_b))  (M=T,N=D,K=16)
// 4 chained WMMAs per tile; bias + double-softplus fused in epilogue.
// ---------------------------------------------------------------------------
__global__ void k_dt_wmma(const float* __restrict__ proj,   // (T, 32)
                          const float* __restrict__ dtw,    // (D, 16)
                          const float* __restrict__ dtb,    // (D)
                          float* __restrict__ dtv) {        // (T, D)
  const int lane = threadIdx.x & 31;
  const int waveId = blockIdx.x * (blockDim.x >> 5) + (threadIdx.x >> 5); // 0..16383
  const int m0 = (waveId >> 7) * 16;
  const int n0 = (waveId & 127) * 16;
  const int l16 = lane & 15;
  const int koff = (lane >> 4) << 1;

  const float* arow = proj + (m0 + l16) * (2 * S);
  const float* brow = dtw + (n0 + l16) * S;

  v8f c = {};
#pragma unroll
  for (int k0 = 0; k0 < S; k0 += 4) {
    v2f a = *(const v2f*)(arow + k0 + koff);
    v2f b = *(const v2f*)(brow + k0 + koff);
    c = __builtin_amdgcn_wmma_f32_16x16x4_f32(false, a, false, b,
                                              (short)0, c, false, false);
  }

  const int col = n0 + l16;
  const float bias = dtb[col];
  const int rbase = m0 + (lane >> 4) * 8;
#pragma unroll
  for (int r = 0; r < 8; ++r) {
    const float v = softplusf(softplusf(c[r] + bias));
    dtv[(rbase + r) * D + col] = v;
  }
}

// ---------------------------------------------------------------------------
// Kernel 4: sequential scan over t + output reduction, LDS double-buffered.
// Lane = one (d, s) chain. Block = 16 channels; 128 blocks x 256 threads.
// Chunks of CH=64 timesteps are async-copied into LDS one chunk ahead
// (global_load_async_to_lds_b128 + s_wait_asynccnt), so the serial loop
// touches only LDS. 32 KB LDS per block (double buffer).
// ---------------------------------------------------------------------------
constexpr int CH = 64;        // timesteps per chunk
constexpr int NC = T / CH;    // 32 chunks

__global__ void k_scan(const float* __restrict__ xc,
                       const float* __restrict__ proj,
                       const float* __restrict__ dtv,
                       const float* __restrict__ Dp,
                       float* __restrict__ y) {
  __shared__ __attribute__((aligned(16))) float sproj[2][CH * 2 * S]; // 8 KB ea
  __shared__ __attribute__((aligned(16))) float sdt[2][CH * 16];     // 4 KB ea
  __shared__ __attribute__((aligned(16))) float sxc[2][CH * 16];     // 4 KB ea

  const int tid = threadIdx.x;
  const int lane = tid & 31;
  const int wave = tid >> 5;
  const int s = lane & 15;
  const int half = lane >> 4;
  const int dl = wave * 2 + half;           // 0..15 channel-in-block
  const int d0 = blockIdx.x * 16;
  const int d = d0 + dl;

  const float As = -(float)(s + 1);
  const float Dd = Dp[d];

  // stage chunk c into buffer b: 4 x b128 transfers per thread
  auto stage = [&](int c, int b) {
    const int t0 = c * CH;
#pragma unroll
    for (int j = 0; j < 2; ++j) {
      const int e = tid + j * 256;          // 512 float4s of proj
      cp16(proj + (size_t)t0 * (2 * S) + e * 4, &sproj[b][e * 4]);
    }
    const int r = tid >> 2;                 // 64 rows x 4 float4s
    const int p = (tid & 3) * 4;
    cp16(dtv + (size_t)(t0 + r) * D + d0 + p, &sdt[b][tid * 4]);
    cp16(xc + (size_t)(t0 + r) * D + d0 + p, &sxc[b][tid * 4]);
  };

  stage(0, 0);
  cp_wait_all();
  __syncthreads();

  float h = 0.0f;
  for (int c = 0; c < NC; ++c) {
    const int b = c & 1;
    if (c + 1 < NC) stage(c + 1, b ^ 1);    // prefetch next chunk while computing

    const int t0 = c * CH;
#pragma unroll 4
    for (int tl = 0; tl < CH; ++tl) {
      const float dt = sdt[b][tl * 16 + dl];      // LDS broadcast
      const float xv = sxc[b][tl * 16 + dl];
      const float Bs = sproj[b][tl * (2 * S) + s];
      const float Cs = sproj[b][tl * (2 * S) + S + s];

      float dA = __expf(dt * As);
      dA = fmaxf(dA, TINYV);
      const float dBx = fmaxf(dt * Bs * xv, TINYV);
      h = fmaf(dA, h, dBx);

      float yp = Cs * h;
      yp += __shfl_xor(yp, 8);
      yp += __shfl_xor(yp, 4);
      yp += __shfl_xor(yp, 2);
      yp += __shfl_xor(yp, 1);
      if (s == 0) y[(t0 + tl) * D + d] = fmaf(Dd, xv, yp);
    }

    cp_wait_all();     // our async writes for chunk c+1 have landed
    __syncthreads();   // everyone's writes visible before next chunk
  }
}

// ---------------------------------------------------------------------------
extern "C" void kernel_launch(void* const* d_in, const int* in_sizes, int n_in,
                              void* d_out, int out_size, void* d_ws, size_t ws_size,
                              hipStream_t stream) {
  const float* x       = (const float*)d_in[0];
  const float* conv_w  = (const float*)d_in[1];
  const float* conv_b  = (const float*)d_in[2];
  const float* xproj_w = (const float*)d_in[3];
  const float* dt_w    = (const float*)d_in[4];
  const float* dt_b    = (const float*)d_in[5];
  const float* Dp      = (const float*)d_in[6];
  float* y = (float*)d_out;

  float* ws   = (float*)d_ws;
  float* xc   = ws;                           // T*D
  float* proj = xc + (size_t)T * D;           // T*32
  float* dtv  = proj + (size_t)T * 2 * S;     // T*D

  k_conv_silu<<<(T * D) / 256, 256, 0, stream>>>(x, conv_w, conv_b, xc);
  k_proj_wmma<<<32, 256, 0, stream>>>(xc, xproj_w, proj);     // 256 waves
  k_dt_wmma<<<2048, 256, 0, stream>>>(proj, dt_w, dt_b, dtv); // 16384 waves
  k_scan<<<D / 16, 256, 0, stream>>>(xc, proj, dtv, Dp, y);   // 1024 waves
}